// MS_SSIM_Loss_7962869367410
// MI455X (gfx1250) — compile-verified
//
#include <hip/hip_runtime.h>

// ---------------------------------------------------------------------------
// MS-SSIM loss, MI455X (gfx1250).  Separable 11-tap Gaussian convolution done
// as banded matmuls on V_WMMA_F32_16X16X4_F32 (fp32 matrix pipe), fully fused:
//   sigmoid -> (X,Y,X^2,Y^2,XY) -> hconv (WMMA) -> vconv (WMMA) -> ssim/cs
//   -> per-(b,c) reduction, per pyramid level.  Avg-pool kernels build the
// pyramid (sigmoid fused into level-0 downsample so sigmoid(logits) is never
// written to HBM).  Workload is HBM-bandwidth bound; this layout reads each
// input once (+2.6x halo overlap at conv tiles) and writes only the pyramid.
// ---------------------------------------------------------------------------

typedef float v2f __attribute__((ext_vector_type(2)));
typedef float v8f __attribute__((ext_vector_type(8)));

// D = A(16x4) * B(4x16) + C, fp32.  8-arg pattern per probe doc.
#define WMMA4(A, B, C) \
  __builtin_amdgcn_wmma_f32_16x16x4_f32(false, (A), false, (B), (short)0, (C), false, false)

// 11-tap Gaussian, sigma=1.5, normalized (symmetric)
__device__ __constant__ float c_gw[11] = {
    1.0283840e-03f, 7.5988172e-03f, 3.6000773e-02f, 1.0936069e-01f,
    2.1300553e-01f, 2.6601172e-01f, 2.1300553e-01f, 1.0936069e-01f,
    3.6000773e-02f, 7.5988172e-03f, 1.0283840e-03f};

__device__ __forceinline__ float sigmoidf(float x) {
  return 1.0f / (1.0f + __expf(-x));
}

// ---------------------------------------------------------------------------
// Fused SSIM level kernel: one wave (32 threads) per 16x16 output tile.
// LDS: 32x28 input tiles for X and Y; 5 x 28x16 horizontal-conv stash.
// ---------------------------------------------------------------------------
template <int SIG>
__global__ void __launch_bounds__(32)
ssim_level_kernel(const float* __restrict__ X, const float* __restrict__ Y,
                  int H, int W, int tiles_x,
                  float* __restrict__ acc_s, float* __restrict__ acc_c) {
  __shared__ float sX[32][28];
  __shared__ float sY[32][28];
  __shared__ float sH[5][28][16];

  const int lane = threadIdx.x;
  const int n    = lane & 15;   // N index (cols) / M index for A operands
  const int hi   = lane >> 4;   // half-wave selector (K offset +2)
  const int img  = blockIdx.y;
  const int tx   = blockIdx.x % tiles_x;
  const int ty   = blockIdx.x / tiles_x;
  const int oi0  = ty * 16, oj0 = tx * 16;
  const size_t base = (size_t)img * (size_t)H * (size_t)W;

  // ---- load 32x28 input tiles (coords clamped; garbage cols >=26 are
  // multiplied by zero weight entries) --------------------------------------
  for (int idx = lane; idx < 32 * 28; idx += 32) {
    int r = idx / 28, c = idx % 28;
    int gi = oi0 + r; if (gi > H - 1) gi = H - 1;
    int gj = oj0 + c; if (gj > W - 1) gj = W - 1;
    size_t o = base + (size_t)gi * W + gj;
    float xv = X[o];
    if (SIG) xv = sigmoidf(xv);
    sX[r][c] = xv;
    sY[r][c] = Y[o];
  }
  __syncthreads();

  // ---- banded Gaussian weight chunks.  Wb[k][n] = w[k-n] (B layout) and
  // Wv[m][k] = w[k-m] (A layout) are transposes; the A and B VGPR layouts are
  // transposes of each other, so ONE register array serves both passes. -----
  v2f wba[7];
#pragma unroll
  for (int c = 0; c < 7; c++) {
    int k0 = 4 * c + 2 * hi;
    int d0 = k0 - n;
    int d1 = d0 + 1;
    float v0 = (d0 >= 0 && d0 <= 10 && k0 < 26) ? c_gw[d0] : 0.0f;
    float v1 = (d1 >= 0 && d1 <= 10 && (k0 + 1) < 26) ? c_gw[d1] : 0.0f;
    v2f t = {v0, v1};
    wba[c] = t;
  }

  // ---- horizontal pass: H_g(16x16) = Tile_g(16x28) x Wb(28x16) ------------
  v8f hacc[2][5] = {};
#pragma unroll
  for (int g = 0; g < 2; g++) {
    int row = g * 16 + n;  // A layout: m = lane&15
#pragma unroll
    for (int c = 0; c < 7; c++) {
      int k0 = 4 * c + 2 * hi;
      float x0 = sX[row][k0], x1 = sX[row][k0 + 1];
      float y0 = sY[row][k0], y1 = sY[row][k0 + 1];
      v2f aX  = {x0, x1};
      v2f aY  = {y0, y1};
      v2f aXX = {x0 * x0, x1 * x1};
      v2f aYY = {y0 * y0, y1 * y1};
      v2f aXY = {x0 * y0, x1 * y1};
      hacc[g][0] = WMMA4(aX,  wba[c], hacc[g][0]);
      hacc[g][1] = WMMA4(aY,  wba[c], hacc[g][1]);
      hacc[g][2] = WMMA4(aXX, wba[c], hacc[g][2]);
      hacc[g][3] = WMMA4(aYY, wba[c], hacc[g][3]);
      hacc[g][4] = WMMA4(aXY, wba[c], hacc[g][4]);
    }
  }

  // ---- stash H to LDS (rows 0..25 valid; 26,27 zero-padded) ---------------
#pragma unroll
  for (int q = 0; q < 5; q++) sH[q][26 + hi][n] = 0.0f;
#pragma unroll
  for (int g = 0; g < 2; g++)
#pragma unroll
    for (int q = 0; q < 5; q++)
#pragma unroll
      for (int v = 0; v < 8; v++) {
        int r = g * 16 + v + 8 * hi;  // C/D layout: m = v + 8*(lane>=16)
        if (r < 26) sH[q][r][n] = hacc[g][q][v];
      }
  __syncthreads();

  // ---- vertical pass: Out(16x16) = Wv(16x28) x H(28x16) -------------------
  v8f vacc[5] = {};
#pragma unroll
  for (int c = 0; c < 7; c++) {
    int k0 = 4 * c + 2 * hi;  // B layout row index
#pragma unroll
    for (int q = 0; q < 5; q++) {
      v2f b = {sH[q][k0][n], sH[q][k0 + 1][n]};
      vacc[q] = WMMA4(wba[c], b, vacc[q]);
    }
  }

  // ---- SSIM / CS pointwise + in-tile reduction ----------------------------
  const float C1 = 1.0e-4f, C2 = 9.0e-4f;
  float ssum = 0.0f, csum = 0.0f;
#pragma unroll
  for (int v = 0; v < 8; v++) {
    int oi = oi0 + v + 8 * hi;
    int oj = oj0 + n;
    if (oi < H - 10 && oj < W - 10) {
      float mu1 = vacc[0][v], mu2 = vacc[1][v];
      float exx = vacc[2][v], eyy = vacc[3][v], exy = vacc[4][v];
      float m11 = mu1 * mu1, m22 = mu2 * mu2, m12 = mu1 * mu2;
      float s1 = exx - m11, s2 = eyy - m22, s12 = exy - m12;
      float cs = (2.0f * s12 + C2) / (s1 + s2 + C2);
      float ss = ((2.0f * m12 + C1) / (m11 + m22 + C1)) * cs;
      ssum += ss;
      csum += cs;
    }
  }
  // wave32 butterfly reduction
  for (int m = 16; m >= 1; m >>= 1) {
    ssum += __shfl_xor(ssum, m, 32);
    csum += __shfl_xor(csum, m, 32);
  }
  if (lane == 0) {
    atomicAdd(&acc_s[img], ssum);
    atomicAdd(&acc_c[img], csum);
  }
}

// ---------------------------------------------------------------------------
// 2x2 average pool (stride 2).  SIG=1 applies sigmoid to the inputs (level-0
// X pyramid path, so sigmoid(logits) never round-trips through HBM).
// ---------------------------------------------------------------------------
template <int SIG>
__global__ void __launch_bounds__(256)
downsample_kernel(const float* __restrict__ in, float* __restrict__ out,
                  int Wi, int Ho, int Wo, long total) {
  long gid = (long)blockIdx.x * blockDim.x + threadIdx.x;
  if (gid >= total) return;
  int wo   = (int)(gid % Wo);
  long t   = gid / Wo;
  int ho   = (int)(t % Ho);
  long img = t / Ho;
  const float* p = in + img * (long)(2 * Ho) * Wi + (long)(2 * ho) * Wi + 2 * wo;
  float a = p[0], b = p[1], c = p[Wi], d = p[Wi + 1];
  if (SIG) { a = sigmoidf(a); b = sigmoidf(b); c = sigmoidf(c); d = sigmoidf(d); }
  out[gid] = 0.25f * (a + b + c + d);
}

__global__ void zero_kernel(float* __restrict__ p, int n) {
  int i = blockIdx.x * blockDim.x + threadIdx.x;
  if (i < n) p[i] = 0.0f;
}

// acc layout: [ssim level 0..4][nimg] then [cs level 0..4][nimg]
__global__ void __launch_bounds__(128)
finalize_kernel(const float* __restrict__ acc, float* __restrict__ out, int nimg) {
  __shared__ float red[128];
  const float wts[5]  = {0.0448f, 0.2856f, 0.3001f, 0.2363f, 0.1333f};
  const float cnts[5] = {252004.0f, 60516.0f, 13924.0f, 2916.0f, 484.0f};
  int t = threadIdx.x;
  float val = 0.0f;
  if (t < nimg) {
    val = 1.0f;
#pragma unroll
    for (int l = 0; l < 5; l++) {
      float s = (l < 4) ? acc[(5 + l) * nimg + t]   // relu(cs) for levels 0..3
                        : acc[l * nimg + t];        // relu(ssim) at last level
      float m = fmaxf(s / cnts[l], 0.0f);
      val *= powf(m, wts[l]);
    }
  }
  red[t] = val;
  __syncthreads();
  for (int s = 64; s > 0; s >>= 1) {
    if (t < s) red[t] += red[t + s];
    __syncthreads();
  }
  if (t == 0) out[0] = 1.0f - red[0] / (float)nimg;
}

// ---------------------------------------------------------------------------
extern "C" void kernel_launch(void* const* d_in, const int* in_sizes, int n_in,
                              void* d_out, int out_size, void* d_ws, size_t ws_size,
                              hipStream_t stream) {
  (void)n_in; (void)out_size; (void)ws_size;
  const float* logits  = (const float*)d_in[0];
  const float* targets = (const float*)d_in[1];
  float* out = (float*)d_out;
  char* ws = (char*)d_ws;

  const int NIMG = in_sizes[0] / (512 * 512);  // B*C = 116
  const int Hs[5] = {512, 256, 128, 64, 32};

  // workspace layout: accumulators (10*NIMG floats), then X1..X4, Y1..Y4
  float* acc = (float*)ws;
  size_t off = 8192;
  float* Xl[5] = {nullptr, nullptr, nullptr, nullptr, nullptr};
  float* Yl[5] = {nullptr, nullptr, nullptr, nullptr, nullptr};
  for (int l = 1; l < 5; l++) {
    Xl[l] = (float*)(ws + off);
    off += (size_t)NIMG * Hs[l] * Hs[l] * sizeof(float);
  }
  for (int l = 1; l < 5; l++) {
    Yl[l] = (float*)(ws + off);
    off += (size_t)NIMG * Hs[l] * Hs[l] * sizeof(float);
  }

  const int nacc = 10 * NIMG;
  zero_kernel<<<(nacc + 255) / 256, 256, 0, stream>>>(acc, nacc);

  for (int l = 0; l < 5; l++) {
    int H = Hs[l];
    int tiles = (H - 10 + 15) / 16;
    dim3 grid(tiles * tiles, NIMG);
    const float* Xp = (l == 0) ? logits  : Xl[l];
    const float* Yp = (l == 0) ? targets : Yl[l];
    float* as = acc + l * NIMG;
    float* ac = acc + (5 + l) * NIMG;
    if (l == 0)
      ssim_level_kernel<1><<<grid, 32, 0, stream>>>(Xp, Yp, H, H, tiles, as, ac);
    else
      ssim_level_kernel<0><<<grid, 32, 0, stream>>>(Xp, Yp, H, H, tiles, as, ac);

    if (l < 4) {
      int Ho = Hs[l + 1];
      long total = (long)NIMG * Ho * Ho;
      int blocks = (int)((total + 255) / 256);
      if (l == 0) {
        downsample_kernel<1><<<blocks, 256, 0, stream>>>(logits,  Xl[1], H, Ho, Ho, total);
        downsample_kernel<0><<<blocks, 256, 0, stream>>>(targets, Yl[1], H, Ho, Ho, total);
      } else {
        downsample_kernel<0><<<blocks, 256, 0, stream>>>(Xl[l], Xl[l + 1], H, Ho, Ho, total);
        downsample_kernel<0><<<blocks, 256, 0, stream>>>(Yl[l], Yl[l + 1], H, Ho, Ho, total);
      }
    }
  }

  finalize_kernel<<<1, 128, 0, stream>>>(acc, out, NIMG);
}